// DMLoss_73297911873696
// MI455X (gfx1250) — compile-verified
//
#include <hip/hip_runtime.h>

typedef float v2f __attribute__((ext_vector_type(2)));
typedef float v8f __attribute__((ext_vector_type(8)));

#define NP     128
#define NG     128
#define TIMEI  5
#define NGI    (NG * TIMEI)   // 640 interpolated gt points
#define NTILES (NGI / 16)     // 40 g-tiles of 16

// One workgroup (256 threads = 8 wave32) per batch element.
//
// Phase 1: gt -> LDS, build interleaved interp points (float2) + |g|^2.
// Phase 2: each wave owns 16 pred rows. V_WMMA_F32_16X16X4_F32 computes
//          score[p,g] = |g|^2 - 2*p.g per 16x16 tile:
//            A = -2*pred (K=2,3 lanes zeroed -> B's K=2,3 rows are
//                multiplied by 0, so B loads are fully unconditional),
//            B = interp points (one ds_load_b64, immediate offset),
//            C = broadcast |g|^2 per column.
//          Fully unrolled 40-tile loop keeps running min/argmin, then a
//          ds_swizzle xor-reduction across each 16-lane N-group (index
//          tie-break preserves jnp.argmin first-occurrence semantics).
// Phase 3: smooth-L1 vs matched points, block reduce, partial[b].
__global__ __launch_bounds__(256) void dm_loss_batch(
    const float* __restrict__ pred,   // [B, NP, 2]
    const float* __restrict__ gt,     // [B, NG, 2]
    float* __restrict__ partial)      // [B]
{
    __shared__ float  s_gxy[NG * 2];  // raw gt, interleaved as in memory
    __shared__ float2 s_ixy[NGI];     // interp points, interleaved
    __shared__ float  s_g2[NGI];      // |g|^2
    __shared__ int    s_arg[NP];
    __shared__ float  s_red[256];

    const int b   = blockIdx.x;
    const int tid = threadIdx.x;
    const float* gtb = gt   + (size_t)b * NG * 2;
    const float* prb = pred + (size_t)b * NP * 2;

    // ---- Phase 1: load gt (256 floats, one per thread), interpolate ----
    s_gxy[tid] = gtb[tid];
    __syncthreads();

    for (int i = tid; i < NGI; i += 256) {
        const int n  = i / TIMEI;
        const int t  = i - n * TIMEI;
        const int np = (n + NG - 1) & (NG - 1);      // roll(shift=1)
        const float w = (float)t / (float)TIMEI;     // arange(T)/T in f32
        const float x = s_gxy[2 * n + 0] * w + s_gxy[2 * np + 0] * (1.0f - w);
        const float y = s_gxy[2 * n + 1] * w + s_gxy[2 * np + 1] * (1.0f - w);
        s_ixy[i] = make_float2(x, y);
        s_g2[i]  = x * x + y * y;
    }
    __syncthreads();

    // ---- Phase 2: WMMA scoring + argmin ----
    const int  wave = tid >> 5;
    const int  lane = tid & 31;
    const int  l16  = lane & 15;
    const bool lo   = lane < 16;          // hi lanes carry K=2,3 -> A zeroed
    const int  prow = wave * 16 + l16;

    const float px = prb[2 * prow + 0];   // unconditional, always in-bounds
    const float py = prb[2 * prow + 1];
    v2f a;
    a.x = lo ? -2.0f * px : 0.0f;         // fold the -2 factor into A
    a.y = lo ? -2.0f * py : 0.0f;

    const float2* bbase = &s_ixy[l16];    // valid for ALL lanes (hi-lane data
    const float*  cbase = &s_g2[l16];     // is multiplied by A's zero K=2,3)

    float minv[8];
    int   mini[8];
#pragma unroll
    for (int r = 0; r < 8; ++r) { minv[r] = 3.0e38f; mini[r] = 0; }

#pragma unroll
    for (int tile = 0; tile < NTILES; ++tile) {
        const float2 bm2 = bbase[tile * 16];          // ds_load_b64 imm offset
        const float  cv  = cbase[tile * 16];          // ds_load_b32 imm offset
        v2f bm; bm.x = bm2.x; bm.y = bm2.y;
        v8f c = { cv, cv, cv, cv, cv, cv, cv, cv };   // per-column |g|^2 bias
        // D[p,g] = |g|^2 - 2*p.g  (16x16 tile, f32 exact)
        v8f d = __builtin_amdgcn_wmma_f32_16x16x4_f32(
            /*neg_a=*/false, a, /*neg_b=*/false, bm,
            /*c_mod=*/(short)0, c, /*reuse_a=*/false, /*reuse_b=*/false);
        const int g = tile * 16 + l16;
#pragma unroll
        for (int r = 0; r < 8; ++r) {
            const float v = d[r];
            if (v < minv[r]) { minv[r] = v; mini[r] = g; }  // strict <: first hit
        }
    }

    // xor-reduce across the 16 lanes sharing a D row; masks 8/4/2/1 stay
    // inside each 16-lane group. ds_swizzle group-of-32: {xor<<10 | and=0x1F}.
#pragma unroll
    for (int r = 0; r < 8; ++r) {
        float v = minv[r];
        int   i = mini[r];
#define ARGMIN_STEP(PAT)                                                      \
        {                                                                     \
            const float ov = __int_as_float(                                  \
                __builtin_amdgcn_ds_swizzle(__float_as_int(v), (PAT)));       \
            const int   oi = __builtin_amdgcn_ds_swizzle(i, (PAT));           \
            if (ov < v || (ov == v && oi < i)) { v = ov; i = oi; }            \
        }
        ARGMIN_STEP(0x201F)   // xor 8
        ARGMIN_STEP(0x101F)   // xor 4
        ARGMIN_STEP(0x081F)   // xor 2
        ARGMIN_STEP(0x041F)   // xor 1
#undef ARGMIN_STEP
        if (lane == 0)  s_arg[wave * 16 + r]     = i;  // rows M=0..7
        if (lane == 16) s_arg[wave * 16 + r + 8] = i;  // rows M=8..15
    }
    __syncthreads();

    // ---- Phase 3: smooth-L1 partial sum ----
    float acc = 0.0f;
    if (tid < NP) {
        const float2 m = s_ixy[s_arg[tid]];
        const float dx = fabsf(prb[2 * tid + 0] - m.x);
        const float dy = fabsf(prb[2 * tid + 1] - m.y);
        acc  = (dx < 1.0f) ? 0.5f * dx * dx : dx - 0.5f;
        acc += (dy < 1.0f) ? 0.5f * dy * dy : dy - 0.5f;
    }
    s_red[tid] = acc;
    __syncthreads();
    for (int s = 128; s >= 1; s >>= 1) {
        if (tid < s) s_red[tid] += s_red[tid + s];
        __syncthreads();
    }
    if (tid == 0) partial[b] = s_red[0];
}

__global__ __launch_bounds__(256) void dm_loss_finalize(
    const float* __restrict__ partial, float* __restrict__ out,
    int nb, float inv_count)
{
    __shared__ float s[256];
    const int tid = threadIdx.x;
    float a = 0.0f;
    for (int i = tid; i < nb; i += 256) a += partial[i];
    s[tid] = a;
    __syncthreads();
    for (int st = 128; st >= 1; st >>= 1) {
        if (tid < st) s[tid] += s[tid + st];
        __syncthreads();
    }
    if (tid == 0) out[0] = s[0] * inv_count;
}

extern "C" void kernel_launch(void* const* d_in, const int* in_sizes, int n_in,
                              void* d_out, int out_size, void* d_ws, size_t ws_size,
                              hipStream_t stream) {
    // inputs: 0 = init_polys (unused by forward value), 1 = pred_poly, 2 = gt_polys
    const float* pred = (const float*)d_in[1];
    const float* gt   = (const float*)d_in[2];
    float* partial    = (float*)d_ws;           // B floats of scratch
    float* out        = (float*)d_out;

    const int B = in_sizes[1] / (NP * 2);       // 512 with the reference shapes

    dm_loss_batch<<<B, 256, 0, stream>>>(pred, gt, partial);

    const float inv_count = 1.0f / (float)(B * NP * 2);
    dm_loss_finalize<<<1, 256, 0, stream>>>(partial, out, B, inv_count);
}